// BiMPM_893353197848
// MI455X (gfx1250) — compile-verified
//
#include <hip/hip_runtime.h>

#define BB 128
#define LL 48
#define EE 300
#define HH 128
#define G4H 512
#define MM 20
#define ADIM 100
#define OUTC 368
#define EPSF 1e-6f
#define NEGF -1e9f

typedef __attribute__((ext_vector_type(2))) float v2f;
typedef __attribute__((ext_vector_type(8))) float v8f;
typedef unsigned int u32;
typedef u32 __attribute__((ext_vector_type(4))) v4u;
typedef int __attribute__((ext_vector_type(4))) v4i;
typedef int __attribute__((ext_vector_type(8))) v8i;

#if defined(__has_builtin)
#if __has_builtin(__builtin_amdgcn_tensor_load_to_lds) && __has_builtin(__builtin_amdgcn_s_wait_tensorcnt)
#define HAVE_TDM 1
#endif
#endif

static __device__ __forceinline__ v8f wmma_f32(v2f a, v2f b, v8f c) {
  // 8 args: (neg_a, A, neg_b, B, c_mod, C, reuse_a, reuse_b)
  return __builtin_amdgcn_wmma_f32_16x16x4_f32(false, a, false, b, (short)0, c, false, false);
}

static __device__ __forceinline__ float sigf(float x) {
  return 1.0f / (1.0f + __expf(-x));
}

#ifdef HAVE_TDM
// Issue a TDM 2D tile load: 16 rows x 512 f32, row stride LL*512 elements,
// destination = LDS byte offset lds_off. (D# per CDNA5 ISA)
// amdgpu-toolchain (clang-23) builtin: 6 args (g0 x4, g1 x8, x4, x4, x8, cpol).
static __device__ __forceinline__ void tdm_load_16x512(const float* gsrc, u32 lds_off) {
  unsigned long long ga = (unsigned long long)(uintptr_t)gsrc;
  v4u g0;
  g0[0] = 1u;                                            // count=1 (valid), no gather
  g0[1] = lds_off;                                       // lds_addr (bytes)
  g0[2] = (u32)(ga & 0xFFFFFFFFu);                       // global_addr[31:0]
  g0[3] = (u32)((ga >> 32) & 0x01FFFFFFu) | (2u << 30);  // global_addr[56:32] | type=2
  v8i g1;
  g1[0] = (int)(2u << 16);                   // workgroup_mask=0, data_size=2 (4 bytes)
  g1[1] = (int)((512u & 0xFFFFu) << 16);     // tensor_dim0[15:0]=512
  g1[2] = (int)((16u & 0xFFFFu) << 16);      // tensor_dim0[31:16]=0, tensor_dim1[15:0]=16
  g1[3] = (int)(512u << 16);                 // tensor_dim1[31:16]=0, tile_dim0=512
  g1[4] = (int)16u;                          // tile_dim1=16, tile_dim2=0
  g1[5] = (int)(LL * 512u);                  // tensor_dim0_stride lo32 = 24576
  g1[6] = 0;                                 // stride hi, dim1_stride lo
  g1[7] = 0;
  v4i z4 = {0, 0, 0, 0};
  v8i z8 = {0, 0, 0, 0, 0, 0, 0, 0};
  __builtin_amdgcn_tensor_load_to_lds(g0, g1, z4, z4, z8, 0);
}
#endif

// ---------------- K1: embedding gather (x1, x2, and per-row reversed+masked x1r, x2r)
__global__ void k_gather(const int* __restrict__ q1_tok, const int* __restrict__ q2_tok,
                         const int* __restrict__ q1_len, const int* __restrict__ q2_len,
                         const float* __restrict__ emb,
                         float* __restrict__ x1, float* __restrict__ x2,
                         float* __restrict__ x1r, float* __restrict__ x2r) {
  int bl = blockIdx.x;            // B*L
  int b = bl / LL, l = bl % LL;
  int len1 = q1_len[b], len2 = q2_len[b];
  int t1 = q1_tok[bl], t2 = q2_tok[bl];
  int r1 = (l < len1) ? q1_tok[b * LL + (len1 - 1 - l)] : -1;
  int r2 = (l < len2) ? q2_tok[b * LL + (len2 - 1 - l)] : -1;
  for (int e = threadIdx.x; e < EE; e += blockDim.x) {
    x1[(size_t)bl * EE + e] = emb[(size_t)t1 * EE + e];
    x2[(size_t)bl * EE + e] = emb[(size_t)t2 * EE + e];
    x1r[(size_t)bl * EE + e] = (r1 >= 0) ? emb[(size_t)r1 * EE + e] : 0.0f;
    x2r[(size_t)bl * EE + e] = (r2 >= 0) ? emb[(size_t)r2 * EE + e] : 0.0f;
  }
}

// ---------------- K2: input GEMM  gx[seq] = X @ W_ih^T + (b_ih + b_hh)
// M = B*L = 6144, N = 512, K = 300. One wave per 16x64 strip: A fragment reused 4x.
__global__ __launch_bounds__(256) void k_ingemm(
    const float* __restrict__ x1, const float* __restrict__ x1r,
    const float* __restrict__ x2, const float* __restrict__ x2r,
    const float* __restrict__ wf, const float* __restrict__ wb,
    const float* __restrict__ bihf, const float* __restrict__ bhhf,
    const float* __restrict__ bihb, const float* __restrict__ bhhb,
    float* __restrict__ gx) {
  int seq = blockIdx.y;  // 0:q1_fw 1:q1_bw 2:q2_fw 3:q2_bw
  const float* X = (seq == 0) ? x1 : (seq == 1) ? x1r : (seq == 2) ? x2 : x2r;
  const float* W = (seq == 0 || seq == 2) ? wf : wb;          // (512,300) row-major
  const float* bi = (seq == 0 || seq == 2) ? bihf : bihb;
  const float* bh = (seq == 0 || seq == 2) ? bhhf : bhhb;
  int wave = threadIdx.x >> 5, lane = threadIdx.x & 31;
  int task = blockIdx.x * 8 + wave;      // 3072 tasks = 384 m-tiles x 8 n-groups
  int tm = task >> 3, tng = task & 7;
  int m0 = tm * 16, n0 = tng * 64;
  int mrow = lane & 15;
  int kk = (lane >> 4) * 2;
  int rbase = 8 * (lane >> 4);
  const float* Arow = X + (size_t)(m0 + mrow) * EE;
  const float* B0 = W + (size_t)(n0 + mrow) * EE;   // B[k][n] = W[n*EE + k]
  const float* B1 = B0 + (size_t)16 * EE;
  const float* B2 = B0 + (size_t)32 * EE;
  const float* B3 = B0 + (size_t)48 * EE;
  v8f acc0 = {0.f, 0.f, 0.f, 0.f, 0.f, 0.f, 0.f, 0.f};
  v8f acc1 = acc0, acc2 = acc0, acc3 = acc0;
  for (int k = 0; k < EE; k += 4) {
    v2f a, b0q, b1q, b2q, b3q;
    a.x = Arow[k + kk];   a.y = Arow[k + kk + 1];
    b0q.x = B0[k + kk];   b0q.y = B0[k + kk + 1];
    b1q.x = B1[k + kk];   b1q.y = B1[k + kk + 1];
    b2q.x = B2[k + kk];   b2q.y = B2[k + kk + 1];
    b3q.x = B3[k + kk];   b3q.y = B3[k + kk + 1];
    acc0 = wmma_f32(a, b0q, acc0);
    acc1 = wmma_f32(a, b1q, acc1);
    acc2 = wmma_f32(a, b2q, acc2);
    acc3 = wmma_f32(a, b3q, acc3);
  }
  float* G = gx + (size_t)seq * BB * LL * G4H;
#pragma unroll
  for (int j = 0; j < 4; j++) {
    v8f acc = (j == 0) ? acc0 : (j == 1) ? acc1 : (j == 2) ? acc2 : acc3;
    int n = n0 + j * 16 + mrow;
    float biasv = bi[n] + bh[n];
#pragma unroll
    for (int v = 0; v < 8; v++) {
      int r = m0 + rbase + v;
      G[(size_t)r * G4H + n] = acc[v] + biasv;
    }
  }
}

// ---------------- K3: LSTM recurrence. 4 seqs x 8 batch-chunks of 16 rows.
// LDS-resident h/c, WMMA h @ W_hh^T per step; TDM double-buffered prefetch of gx[t].
__global__ __launch_bounds__(256) void k_lstm(
    const float* __restrict__ gx, const float* __restrict__ whf,
    const float* __restrict__ whb, const int* __restrict__ q1_len,
    const int* __restrict__ q2_len, float* __restrict__ hs) {
  __shared__ float h_s[16 * HH];
  __shared__ float c_s[16 * HH];
  __shared__ float g_s[16 * G4H];
#ifdef HAVE_TDM
  __shared__ float gxb[2][16 * G4H];
#endif
  int seq = blockIdx.x >> 3;
  int chunk = blockIdx.x & 7;
  const float* Wh = (seq == 0 || seq == 2) ? whf : whb;  // (512,128) row-major
  const int* lens = (seq < 2) ? q1_len : q2_len;
  bool is_bw = (seq & 1);
  const float* G = gx + (size_t)seq * BB * LL * G4H;
  float* HS = hs + (size_t)seq * BB * LL * HH;
  int tid = threadIdx.x;
  for (int i = tid; i < 16 * HH; i += blockDim.x) { h_s[i] = 0.f; c_s[i] = 0.f; }
  int wave = tid >> 5, lane = tid & 31;
  int mrow = lane & 15;
  int kk = (lane >> 4) * 2;
  int rbase = 8 * (lane >> 4);
  int b0row = chunk * 16;
#ifdef HAVE_TDM
  u32 lds0 = (u32)(uintptr_t)(&gxb[0][0]);
  u32 lds1 = (u32)(uintptr_t)(&gxb[1][0]);
  if (tid < 32) {  // wave 0 drives the TDM pipeline
    tdm_load_16x512(G + ((size_t)b0row * LL + 0) * G4H, lds0);
  }
#endif
  __syncthreads();
  // wave's 4 n-tiles: n0..n0+63 (A fragment reused 4x per k-step)
  int n0 = wave * 64;
  const float* Br0 = Wh + (size_t)(n0 + mrow) * HH;
  const float* Br1 = Br0 + (size_t)16 * HH;
  const float* Br2 = Br0 + (size_t)32 * HH;
  const float* Br3 = Br0 + (size_t)48 * HH;
  for (int t = 0; t < LL; t++) {
#ifdef HAVE_TDM
    if (tid < 32 && t + 1 < LL) {
      tdm_load_16x512(G + ((size_t)b0row * LL + (t + 1)) * G4H, (t & 1) ? lds0 : lds1);
    }
#endif
    // phase A: g_s = h_s @ Wh^T  (16 x 512); this wave: 16x64 strip
    {
      const float* Arow = h_s + mrow * HH;
      v8f acc0 = {0.f, 0.f, 0.f, 0.f, 0.f, 0.f, 0.f, 0.f};
      v8f acc1 = acc0, acc2 = acc0, acc3 = acc0;
      for (int k = 0; k < HH; k += 4) {
        v2f a, b0q, b1q, b2q, b3q;
        a.x = Arow[k + kk];   a.y = Arow[k + kk + 1];
        b0q.x = Br0[k + kk];  b0q.y = Br0[k + kk + 1];
        b1q.x = Br1[k + kk];  b1q.y = Br1[k + kk + 1];
        b2q.x = Br2[k + kk];  b2q.y = Br2[k + kk + 1];
        b3q.x = Br3[k + kk];  b3q.y = Br3[k + kk + 1];
        acc0 = wmma_f32(a, b0q, acc0);
        acc1 = wmma_f32(a, b1q, acc1);
        acc2 = wmma_f32(a, b2q, acc2);
        acc3 = wmma_f32(a, b3q, acc3);
      }
#pragma unroll
      for (int j = 0; j < 4; j++) {
        v8f acc = (j == 0) ? acc0 : (j == 1) ? acc1 : (j == 2) ? acc2 : acc3;
        int n = n0 + j * 16 + mrow;
#pragma unroll
        for (int v = 0; v < 8; v++) g_s[(rbase + v) * G4H + n] = acc[v];
      }
    }
#ifdef HAVE_TDM
    if (tid < 32) {
      // TDM ops from one wave complete in order: <=1 outstanding means tile t landed.
      if (t + 1 < LL) __builtin_amdgcn_s_wait_tensorcnt(1);
      else            __builtin_amdgcn_s_wait_tensorcnt(0);
    }
#endif
    __syncthreads();
    // phase B: gates (PyTorch order i,f,g,o), update c/h, emit output
    for (int i = tid; i < 16 * HH; i += blockDim.x) {
      int blr = i >> 7;
      int hh = i & 127;
      int b = b0row + blr;
#ifdef HAVE_TDM
      const float* Grow = &gxb[t & 1][blr * G4H];
#else
      const float* Grow = G + ((size_t)b * LL + t) * G4H;
#endif
      float gi = g_s[blr * G4H + hh]       + Grow[hh];
      float gf = g_s[blr * G4H + 128 + hh] + Grow[128 + hh];
      float gg = g_s[blr * G4H + 256 + hh] + Grow[256 + hh];
      float go = g_s[blr * G4H + 384 + hh] + Grow[384 + hh];
      float c = sigf(gf) * c_s[i] + sigf(gi) * tanhf(gg);
      float h = sigf(go) * tanhf(c);
      c_s[i] = c;
      h_s[i] = h;
      int len = lens[b];
      if (t < len) {
        int ot = is_bw ? (len - 1 - t) : t;
        HS[((size_t)b * LL + ot) * HH + hh] = h;
      } else {
        HS[((size_t)b * LL + t) * HH + hh] = 0.f;
      }
    }
    __syncthreads();
  }
}

// ---------------- K4: gather last q2 states
__global__ void k_last(const float* __restrict__ hs, const int* __restrict__ q2_len,
                       float* __restrict__ lfw, float* __restrict__ lbw) {
  int b = blockIdx.x, h = threadIdx.x;
  int len2 = q2_len[b];
  const float* hs2 = hs + (size_t)2 * BB * LL * HH;
  const float* hs3 = hs + (size_t)3 * BB * LL * HH;
  lfw[b * HH + h] = hs2[((size_t)b * LL + len2 - 1) * HH + h];
  lbw[b * HH + h] = hs3[((size_t)b * LL + len2 - 1) * HH + h];
}

// ---------------- K5: plain L1 norms over H for all 4 hs seqs (for cos-attn)
__global__ void k_l1h(const float* __restrict__ hs, float* __restrict__ l1h) {
  int idx = blockIdx.x * blockDim.x + threadIdx.x;
  if (idx >= 4 * BB * LL) return;
  const float* p = hs + (size_t)idx * HH;
  float s = 0.f;
  for (int h = 0; h < HH; h++) s += fabsf(p[h]);
  l1h[idx] = s;
}

// ---------------- K6: weighted L1 norms  dst[r,m] = sum_h |src[r,h]*w[m,h]|
__global__ void k_normw(const float* __restrict__ src, const float* __restrict__ w,
                        float* __restrict__ dst, int rows) {
  int idx = blockIdx.x * blockDim.x + threadIdx.x;
  if (idx >= rows * MM) return;
  int r = idx / MM, m = idx % MM;
  const float* s = src + (size_t)r * HH;
  const float* ww = w + m * HH;
  float acc = 0.f;
  for (int h = 0; h < HH; h++) acc += fabsf(s[h] * ww[h]);
  dst[idx] = acc;
}

// ---------------- K7: full match -> out cols [0,40)
__global__ void k_full(const float* __restrict__ hs, const float* __restrict__ lfw,
                       const float* __restrict__ lbw, const float* __restrict__ full_w,
                       const float* __restrict__ n1f, const float* __restrict__ n1b,
                       const float* __restrict__ nlf, const float* __restrict__ nlb,
                       float* __restrict__ out) {
  int idx = blockIdx.x * blockDim.x + threadIdx.x;
  if (idx >= 2 * BB * LL * MM) return;
  int dir = idx / (BB * LL * MM);
  int rem = idx % (BB * LL * MM);
  int bl = rem / MM, m = rem % MM;
  int b = bl / LL;
  const float* q1 = hs + (size_t)(dir == 0 ? 0 : 1) * BB * LL * HH + (size_t)bl * HH;
  const float* l2 = (dir == 0 ? lfw : lbw) + (size_t)b * HH;
  const float* ww = full_w + m * HH;
  float num = 0.f;
  for (int h = 0; h < HH; h++) { float w2 = ww[h] * ww[h]; num += q1[h] * l2[h] * w2; }
  float den = ((dir == 0 ? n1f : n1b)[bl * MM + m] + EPSF) *
              ((dir == 0 ? nlf : nlb)[b * MM + m] + EPSF);
  out[(size_t)bl * OUTC + dir * MM + m] = num / den;
}

// ---------------- K8: pool-match factorization: t2[b,m,h] = w[m,h]/L * sum_k x2[b,k,h]/(n2[b,k,m]+eps)
__global__ void k_t2(const float* __restrict__ hs, const float* __restrict__ pool_w,
                     const float* __restrict__ n2, float* __restrict__ t2, int srcseq) {
  int idx = blockIdx.x * blockDim.x + threadIdx.x;
  if (idx >= BB * MM * HH) return;
  int b = idx / (MM * HH);
  int m = (idx / HH) % MM;
  int h = idx % HH;
  const float* src = hs + (size_t)srcseq * BB * LL * HH + (size_t)b * LL * HH;
  float s = 0.f;
  for (int k = 0; k < LL; k++)
    s += src[(size_t)k * HH + h] / (n2[((size_t)b * LL + k) * MM + m] + EPSF);
  t2[idx] = s * pool_w[m * HH + h] * (1.0f / (float)LL);
}

__global__ void k_pool(const float* __restrict__ hs, const float* __restrict__ pool_w,
                       const float* __restrict__ t2, const float* __restrict__ n1,
                       float* __restrict__ out, int srcseq, int colbase) {
  int idx = blockIdx.x * blockDim.x + threadIdx.x;
  if (idx >= BB * LL * MM) return;
  int bl = idx / MM, m = idx % MM;
  int b = bl / LL;
  const float* q = hs + (size_t)srcseq * BB * LL * HH + (size_t)bl * HH;
  const float* ww = pool_w + m * HH;
  const float* tt = t2 + ((size_t)b * MM + m) * HH;
  float s = 0.f;
  for (int h = 0; h < HH; h++) s += q[h] * ww[h] * tt[h];
  out[(size_t)bl * OUTC + colbase + m] = s / (n1[bl * MM + m] + EPSF);
}

// ---------------- K9: cosine attention via f32 WMMA (48x48 per batch, K=128) -> cols [80,176)
__global__ __launch_bounds__(256) void k_cos(const float* __restrict__ hs,
                                             const float* __restrict__ l1h,
                                             const int* __restrict__ q1_len,
                                             const int* __restrict__ q2_len,
                                             float* __restrict__ out) {
  int b = blockIdx.x;
  int wave = threadIdx.x >> 5, lane = threadIdx.x & 31;
  int mrow = lane & 15, kk = (lane >> 4) * 2, rbase = 8 * (lane >> 4);
  int len1 = q1_len[b], len2 = q2_len[b];
  for (int tix = wave; tix < 18; tix += 8) {
    int dir = tix / 9, tt = tix % 9;
    int tl = tt / 3, tk = tt % 3;
    int s1 = dir ? 1 : 0, s2 = dir ? 3 : 2;
    const float* A = hs + ((size_t)s1 * BB * LL + (size_t)b * LL) * HH;
    const float* Bm = hs + ((size_t)s2 * BB * LL + (size_t)b * LL) * HH;
    int l0 = tl * 16, k0 = tk * 16;
    const float* Arow = A + (size_t)(l0 + mrow) * HH;
    const float* Brow = Bm + (size_t)(k0 + mrow) * HH;   // B[k][n] = Bm[n][k]
    v8f acc = {0.f, 0.f, 0.f, 0.f, 0.f, 0.f, 0.f, 0.f};
    for (int k = 0; k < HH; k += 4) {
      v2f a, bq;
      a.x = Arow[k + kk];  a.y = Arow[k + kk + 1];
      bq.x = Brow[k + kk]; bq.y = Brow[k + kk + 1];
      acc = wmma_f32(a, bq, acc);
    }
    const float* n1 = l1h + (size_t)s1 * BB * LL + (size_t)b * LL;
    const float* n2 = l1h + (size_t)s2 * BB * LL + (size_t)b * LL;
    int kc = k0 + mrow;
    int colbase = 80 + dir * 48;
#pragma unroll
    for (int v = 0; v < 8; v++) {
      int l = l0 + rbase + v;
      float mask = (l < len1 && kc < len2) ? 1.f : 0.f;
      out[((size_t)b * LL + l) * OUTC + colbase + kc] =
          acc[v] / ((n1[l] + EPSF) * (n2[kc] + EPSF)) * mask;
    }
  }
}

// ---------------- K10: projections for mult/add attention: dst = hs[seq] @ w^T + bias
__global__ void k_proj(const float* __restrict__ hs, const float* __restrict__ w,
                       const float* __restrict__ bias, float* __restrict__ dst, int seq) {
  int idx = blockIdx.x * blockDim.x + threadIdx.x;
  if (idx >= BB * LL * ADIM) return;
  int bl = idx / ADIM, d = idx % ADIM;
  const float* q = hs + (size_t)seq * BB * LL * HH + (size_t)bl * HH;
  const float* ww = w + d * HH;
  float s = bias[d];
  for (int h = 0; h < HH; h++) s += q[h] * ww[h];
  dst[idx] = s;
}

// ---------------- K11: mult-attn logits via f32 WMMA (48x48 per batch, K=100) -> cols [176,272)
__global__ __launch_bounds__(256) void k_multlog(const float* __restrict__ a0,
                                                 const float* __restrict__ a2,
                                                 const float* __restrict__ a1,
                                                 const float* __restrict__ a3,
                                                 float* __restrict__ out) {
  int b = blockIdx.x;
  int wave = threadIdx.x >> 5, lane = threadIdx.x & 31;
  int mrow = lane & 15, kk = (lane >> 4) * 2, rbase = 8 * (lane >> 4);
  for (int tix = wave; tix < 18; tix += 8) {
    int dir = tix / 9, tt = tix % 9;
    int tl = tt / 3, tk = tt % 3;
    const float* A = (dir ? a1 : a0) + (size_t)b * LL * ADIM;
    const float* Bm = (dir ? a3 : a2) + (size_t)b * LL * ADIM;
    int l0 = tl * 16, k0 = tk * 16;
    const float* Arow = A + (size_t)(l0 + mrow) * ADIM;
    const float* Brow = Bm + (size_t)(k0 + mrow) * ADIM;
    v8f acc = {0.f, 0.f, 0.f, 0.f, 0.f, 0.f, 0.f, 0.f};
    for (int k = 0; k < ADIM; k += 4) {
      v2f a, bq;
      a.x = Arow[k + kk];  a.y = Arow[k + kk + 1];
      bq.x = Brow[k + kk]; bq.y = Brow[k + kk + 1];
      acc = wmma_f32(a, bq, acc);
    }
    int kc = k0 + mrow;
    int colbase = 176 + dir * 48;
#pragma unroll
    for (int v = 0; v < 8; v++) {
      int l = l0 + rbase + v;
      out[((size_t)b * LL + l) * OUTC + colbase + kc] = acc[v];
    }
  }
}

// ---------------- K12: add-attn logits (tanh path) -> cols [272,368)
__global__ void k_addlog(const float* __restrict__ a0, const float* __restrict__ a2,
                         const float* __restrict__ a1, const float* __restrict__ a3,
                         const float* __restrict__ add_dot, float* __restrict__ out) {
  int idx = blockIdx.x * blockDim.x + threadIdx.x;
  if (idx >= 2 * BB * LL * LL) return;
  int dir = idx / (BB * LL * LL);
  int rem = idx % (BB * LL * LL);
  int b = rem / (LL * LL);
  int l = (rem / LL) % LL;
  int k = rem % LL;
  const float* pa = (dir ? a1 : a0) + ((size_t)b * LL + l) * ADIM;
  const float* pc = (dir ? a3 : a2) + ((size_t)b * LL + k) * ADIM;
  float s = 0.f;
  for (int d = 0; d < ADIM; d++) s += add_dot[d] * tanhf(pa[d] + pc[d]);
  out[((size_t)b * LL + l) * OUTC + 272 + dir * 48 + k] = s;
}

// ---------------- K13: post-attn softmax-normalize in place over 4 slices of 48 cols
__global__ void k_postattn(const int* __restrict__ q1_len, const int* __restrict__ q2_len,
                           float* __restrict__ out) {
  int idx = blockIdx.x * blockDim.x + threadIdx.x;
  if (idx >= 4 * BB * LL) return;
  int slice = idx / (BB * LL);
  int bl = idx % (BB * LL);
  int b = bl / LL, l = bl % LL;
  int len1 = q1_len[b], len2 = q2_len[b];
  float* row = out + (size_t)bl * OUTC + 176 + slice * 48;
  float vbuf[LL];
  float mx = -3.4e38f;
  for (int k = 0; k < LL; k++) {
    float v = (k < len2) ? row[k] : NEGF;
    vbuf[k] = v;
    mx = fmaxf(mx, v);
  }
  float sum = 0.f;
  for (int k = 0; k < LL; k++) {
    float e = __expf(vbuf[k] - mx) * ((l < len1 && k < len2) ? 1.f : 0.f);
    vbuf[k] = e;
    sum += e;
  }
  float inv = 1.f / (sum + EPSF);
  for (int k = 0; k < LL; k++) row[k] = vbuf[k] * inv;
}

static inline int cdiv(int a, int b) { return (a + b - 1) / b; }

extern "C" void kernel_launch(void* const* d_in, const int* in_sizes, int n_in,
                              void* d_out, int out_size, void* d_ws, size_t ws_size,
                              hipStream_t stream) {
  (void)in_sizes; (void)n_in; (void)out_size; (void)ws_size;
  const int* q1_tok = (const int*)d_in[0];
  const int* q2_tok = (const int*)d_in[1];
  const int* q1_len = (const int*)d_in[2];
  const int* q2_len = (const int*)d_in[3];
  const float* emb    = (const float*)d_in[4];
  const float* w_ih_f = (const float*)d_in[5];
  const float* w_hh_f = (const float*)d_in[6];
  const float* b_ih_f = (const float*)d_in[7];
  const float* b_hh_f = (const float*)d_in[8];
  const float* w_ih_b = (const float*)d_in[9];
  const float* w_hh_b = (const float*)d_in[10];
  const float* b_ih_b = (const float*)d_in[11];
  const float* b_hh_b = (const float*)d_in[12];
  const float* full_w = (const float*)d_in[13];
  const float* pool_w = (const float*)d_in[14];
  const float* mult_w = (const float*)d_in[15];
  const float* mult_b = (const float*)d_in[16];
  const float* add_w  = (const float*)d_in[17];
  const float* add_b  = (const float*)d_in[18];
  const float* add_dot = (const float*)d_in[19];
  float* out = (float*)d_out;
  float* ws = (float*)d_ws;

  // workspace layout (floats)
  const size_t XS = (size_t)BB * LL * EE;
  float* x1  = ws;            float* x2  = x1 + XS;
  float* x1r = x2 + XS;       float* x2r = x1r + XS;
  float* gx  = x2r + XS;                              // 4*B*L*512
  float* hs  = gx + (size_t)4 * BB * LL * G4H;        // 4*B*L*128
  float* lfw = hs + (size_t)4 * BB * LL * HH;         // B*H
  float* lbw = lfw + (size_t)BB * HH;
  float* l1h = lbw + (size_t)BB * HH;                 // 4*B*L
  float* n1f = l1h + (size_t)4 * BB * LL;             // B*L*M each
  float* n1b = n1f + (size_t)BB * LL * MM;
  float* nlf = n1b + (size_t)BB * LL * MM;            // B*M each
  float* nlb = nlf + (size_t)BB * MM;
  float* np0 = nlb + (size_t)BB * MM;                 // pool norms hs0
  float* np2 = np0 + (size_t)BB * LL * MM;            // hs2
  float* np3 = np2 + (size_t)BB * LL * MM;            // hs3
  float* t2f = np3 + (size_t)BB * LL * MM;            // B*M*H each
  float* t2b = t2f + (size_t)BB * MM * HH;
  float* am0 = t2b + (size_t)BB * MM * HH;            // B*L*AD each
  float* am2 = am0 + (size_t)BB * LL * ADIM;
  float* am1 = am2 + (size_t)BB * LL * ADIM;
  float* am3 = am1 + (size_t)BB * LL * ADIM;
  float* aa0 = am3 + (size_t)BB * LL * ADIM;
  float* aa2 = aa0 + (size_t)BB * LL * ADIM;
  float* aa1 = aa2 + (size_t)BB * LL * ADIM;
  float* aa3 = aa1 + (size_t)BB * LL * ADIM;

  // 1) embedding gather (+reversed, masked)
  k_gather<<<BB * LL, 128, 0, stream>>>(q1_tok, q2_tok, q1_len, q2_len, emb,
                                        x1, x2, x1r, x2r);
  // 2) input GEMMs with bias fold (WMMA f32, 16x64 strips)
  k_ingemm<<<dim3(384, 4), 256, 0, stream>>>(x1, x1r, x2, x2r, w_ih_f, w_ih_b,
                                             b_ih_f, b_hh_f, b_ih_b, b_hh_b, gx);
  // 3) LSTM recurrence (WMMA f32, LDS-resident state, TDM gx prefetch)
  k_lstm<<<32, 256, 0, stream>>>(gx, w_hh_f, w_hh_b, q1_len, q2_len, hs);
  // 4) last q2 states
  k_last<<<BB, HH, 0, stream>>>(hs, q2_len, lfw, lbw);
  // 5) norms
  k_l1h<<<cdiv(4 * BB * LL, 256), 256, 0, stream>>>(hs, l1h);
  k_normw<<<cdiv(BB * LL * MM, 256), 256, 0, stream>>>(hs + (size_t)0 * BB * LL * HH, full_w, n1f, BB * LL);
  k_normw<<<cdiv(BB * LL * MM, 256), 256, 0, stream>>>(hs + (size_t)1 * BB * LL * HH, full_w, n1b, BB * LL);
  k_normw<<<cdiv(BB * MM, 256), 256, 0, stream>>>(lfw, full_w, nlf, BB);
  k_normw<<<cdiv(BB * MM, 256), 256, 0, stream>>>(lbw, full_w, nlb, BB);
  k_normw<<<cdiv(BB * LL * MM, 256), 256, 0, stream>>>(hs + (size_t)0 * BB * LL * HH, pool_w, np0, BB * LL);
  k_normw<<<cdiv(BB * LL * MM, 256), 256, 0, stream>>>(hs + (size_t)2 * BB * LL * HH, pool_w, np2, BB * LL);
  k_normw<<<cdiv(BB * LL * MM, 256), 256, 0, stream>>>(hs + (size_t)3 * BB * LL * HH, pool_w, np3, BB * LL);
  // 6) full match -> cols [0,40)
  k_full<<<cdiv(2 * BB * LL * MM, 256), 256, 0, stream>>>(hs, lfw, lbw, full_w,
                                                          n1f, n1b, nlf, nlb, out);
  // 7) pool match (factored) -> cols [40,80)
  k_t2<<<cdiv(BB * MM * HH, 256), 256, 0, stream>>>(hs, pool_w, np2, t2f, 2);
  k_t2<<<cdiv(BB * MM * HH, 256), 256, 0, stream>>>(hs, pool_w, np3, t2b, 3);
  k_pool<<<cdiv(BB * LL * MM, 256), 256, 0, stream>>>(hs, pool_w, t2f, np0, out, 0, 40);
  k_pool<<<cdiv(BB * LL * MM, 256), 256, 0, stream>>>(hs, pool_w, t2b, np3, out, 3, 60);
  // 8) cosine attention (WMMA) -> cols [80,176)
  k_cos<<<BB, 256, 0, stream>>>(hs, l1h, q1_len, q2_len, out);
  // 9) projections
  k_proj<<<cdiv(BB * LL * ADIM, 256), 256, 0, stream>>>(hs, mult_w, mult_b, am0, 0);
  k_proj<<<cdiv(BB * LL * ADIM, 256), 256, 0, stream>>>(hs, mult_w, mult_b, am2, 2);
  k_proj<<<cdiv(BB * LL * ADIM, 256), 256, 0, stream>>>(hs, mult_w, mult_b, am1, 1);
  k_proj<<<cdiv(BB * LL * ADIM, 256), 256, 0, stream>>>(hs, mult_w, mult_b, am3, 3);
  k_proj<<<cdiv(BB * LL * ADIM, 256), 256, 0, stream>>>(hs, add_w, add_b, aa0, 0);
  k_proj<<<cdiv(BB * LL * ADIM, 256), 256, 0, stream>>>(hs, add_w, add_b, aa2, 2);
  k_proj<<<cdiv(BB * LL * ADIM, 256), 256, 0, stream>>>(hs, add_w, add_b, aa1, 1);
  k_proj<<<cdiv(BB * LL * ADIM, 256), 256, 0, stream>>>(hs, add_w, add_b, aa3, 3);
  // 10) mult logits (WMMA) -> cols [176,272)
  k_multlog<<<BB, 256, 0, stream>>>(am0, am2, am1, am3, out);
  // 11) add logits -> cols [272,368)
  k_addlog<<<cdiv(2 * BB * LL * LL, 256), 256, 0, stream>>>(aa0, aa2, aa1, aa3, add_dot, out);
  // 12) post-attn normalize in place over the 4 attention slices
  k_postattn<<<cdiv(4 * BB * LL, 256), 256, 0, stream>>>(q1_len, q2_len, out);
}